// HamiltonianPotential_33363305955950
// MI455X (gfx1250) — compile-verified
//
#include <hip/hip_runtime.h>
#include <math.h>

#define Bc 2
#define Nc 192
#define Kc 16
#define Ac 8
#define Vc 32000
#define EPS_KL 1e-8f
#define EPS_AL 1e-6f

typedef __attribute__((ext_vector_type(2))) float v2f;
typedef __attribute__((ext_vector_type(8))) float v8f;

#define V8Z {0.f,0.f,0.f,0.f,0.f,0.f,0.f,0.f}

// Wave-local LDS ordering: CDNA5 services DS requests from one wave in order
// and wave32 lanes share one instruction stream, so a wavefront-scope fence
// (compiler ordering only, no s_barrier) is sufficient for intra-wave
// producer->consumer through wave-private LDS buffers.
__device__ __forceinline__ void wsync() {
  __builtin_amdgcn_fence(__ATOMIC_ACQ_REL, "wavefront");
}

// ---------- WMMA fp32 16x16 helpers (V_WMMA_F32_16X16X4_F32) ----------
// A 16x4 slice (cols k0..k0+3) of a row-major 16x16 matrix.
// ISA layout: VGPR0 = K(k0) lanes0-15 / K(k0+2) lanes16-31; VGPR1 = k0+1 / k0+3.
__device__ __forceinline__ v2f frag_a_rm(const float* M, int k0, int lane) {
  int h = (lane >> 4) & 1;
  int m = lane & 15;
  v2f r;
  r.x = M[m * 16 + k0 + 2 * h];
  r.y = M[m * 16 + k0 + 2 * h + 1];
  return r;
}
// B 4x16 slice (rows k0..k0+3) of a row-major 16x16 matrix.
__device__ __forceinline__ v2f frag_b_rm(const float* M, int k0, int lane) {
  int h = (lane >> 4) & 1;
  int n = lane & 15;
  v2f r;
  r.x = M[(k0 + 2 * h) * 16 + n];
  r.y = M[(k0 + 2 * h + 1) * 16 + n];
  return r;
}
// B = M^T  (B[k][n] = M[n][k])
__device__ __forceinline__ v2f frag_b_tr(const float* M, int k0, int lane) {
  int h = (lane >> 4) & 1;
  int n = lane & 15;
  v2f r;
  r.x = M[n * 16 + k0 + 2 * h];
  r.y = M[n * 16 + k0 + 2 * h + 1];
  return r;
}

__device__ __forceinline__ v8f wmma4(v2f a, v2f b, v8f c) {
  return __builtin_amdgcn_wmma_f32_16x16x4_f32(false, a, false, b, (short)0, c,
                                               false, false);
}

// D = A(16x16) * B(16x16), both row-major, via 4 chained k=4 WMMAs.
__device__ __forceinline__ v8f mm16(const float* A, const float* B, int lane) {
  v8f c = V8Z;
#pragma unroll
  for (int k0 = 0; k0 < 16; k0 += 4) {
    c = wmma4(frag_a_rm(A, k0, lane), frag_b_rm(B, k0, lane), c);
  }
  return c;
}
// D = A * B^T
__device__ __forceinline__ v8f mm16_bt(const float* A, const float* B, int lane) {
  v8f c = V8Z;
#pragma unroll
  for (int k0 = 0; k0 < 16; k0 += 4) {
    c = wmma4(frag_a_rm(A, k0, lane), frag_b_tr(B, k0, lane), c);
  }
  return c;
}

// C/D layout: VGPR r, lanes0-15 -> (M=r, N=lane); lanes16-31 -> (M=r+8, N=lane-16)
__device__ __forceinline__ void store_c(const v8f& c, float* M, int lane) {
  int h = (lane >> 4) & 1;
  int n = lane & 15;
#pragma unroll
  for (int r = 0; r < 8; ++r) M[(r + 8 * h) * 16 + n] = c[r];
}

// ---------- lane-cooperative 16x16 Cholesky (in-place, lower) ----------
// Matrix M is wave-private LDS; wave-local fences are sufficient.
__device__ __forceinline__ void chol16(float* M, int lane) {
  for (int k = 0; k < 16; ++k) {
    if (lane == 0) M[k * 16 + k] = sqrtf(fmaxf(M[k * 16 + k], 1e-30f));
    wsync();
    float lkk = M[k * 16 + k];
    if (lane > k && lane < 16) M[lane * 16 + k] /= lkk;
    wsync();
    if (lane > k && lane < 16) {
      float ljk = M[lane * 16 + k];
      for (int c = k + 1; c <= lane; ++c) M[lane * 16 + c] -= ljk * M[c * 16 + k];
    }
    wsync();
  }
}

// =====================================================================
// Kernel 1: per-node. expm(+/-Phi) via WMMA, ld_q, self-KL.
// 1 wave per node, 4 waves/block, wave-private LDS (no block barriers).
// =====================================================================
#define NODE_LDS (4 * 256 + 32)
__global__ void __launch_bounds__(128)
node_kernel(const float* __restrict__ mu, const float* __restrict__ Sigma,
            const float* __restrict__ phi, const float* __restrict__ mu_prior,
            const float* __restrict__ Sigma_prior,
            const float* __restrict__ Gn, float* __restrict__ wsEp,
            float* __restrict__ wsEn, float* __restrict__ wsLdq,
            float* __restrict__ out) {
  __shared__ float lds[4 * NODE_LDS];
  const int wid = threadIdx.x >> 5;
  const int lane = threadIdx.x & 31;
  float* T = lds + wid * NODE_LDS;   // scaled Phi
  float* Mc = T + 256;               // Horner accumulator / result
  float* Pr = Mc + 256;              // scratch matrix
  float* S = Pr + 256;               // scratch matrix (chol of Sp)
  float* scr = S + 256;              // 32-float scratch

  const int node = blockIdx.x * 4 + wid;  // 0..383
  const int bb = node / Nc;
  const int e0 = lane * 8;

  // Phi = sum_a phi[a] * G[a], 8 entries per lane held in registers.
  float ph[8];
#pragma unroll
  for (int q = 0; q < 8; ++q) {
    int e = e0 + q;
    float s = 0.f;
    for (int a = 0; a < Ac; ++a) s += phi[node * Ac + a] * Gn[a * 256 + e];
    ph[q] = s;
  }

  // Ep = expm(Phi), En = expm(-Phi): scale by 1/4, Horner Taylor order 7,
  // then 2 squarings.  All matmuls on the WMMA path.
  for (int pass = 0; pass < 2; ++pass) {
    float sgn = (pass == 0) ? 0.25f : -0.25f;
#pragma unroll
    for (int q = 0; q < 8; ++q) {
      int e = e0 + q;
      T[e] = sgn * ph[q];
      Mc[e] = ((e >> 4) == (e & 15)) ? 1.f : 0.f;  // R = I
    }
    wsync();
    for (int k = 7; k >= 1; --k) {  // R = I + (T*R)/k
      v8f c = mm16(T, Mc, lane);
      wsync();
      float inv = 1.f / (float)k;
      int h = (lane >> 4) & 1, n = lane & 15;
#pragma unroll
      for (int r = 0; r < 8; ++r) {
        int row = r + 8 * h;
        Mc[row * 16 + n] = c[r] * inv + ((row == n) ? 1.f : 0.f);
      }
      wsync();
    }
    for (int sq = 0; sq < 2; ++sq) {  // R = R*R twice
      v8f c = mm16(Mc, Mc, lane);
      wsync();
      store_c(c, Mc, lane);
      wsync();
    }
    float* dst = ((pass == 0) ? wsEp : wsEn) + node * 256;
#pragma unroll
    for (int q = 0; q < 8; ++q) dst[e0 + q] = Mc[e0 + q];
    wsync();
  }

  // ld_q = 2*sum log(diag(chol(Sigma + 1e-6 I)) + 1e-6)
#pragma unroll
  for (int q = 0; q < 8; ++q) {
    int e = e0 + q;
    Pr[e] = Sigma[node * 256 + e] + (((e >> 4) == (e & 15)) ? EPS_AL : 0.f);
  }
  wsync();
  chol16(Pr, lane);
  if (lane == 0) {
    float ld = 0.f;
    for (int r = 0; r < 16; ++r) ld += logf(Pr[r * 17] + EPS_AL);
    wsLdq[node] = 2.f * ld;
  }
  wsync();

  // ---- self KL: chol(Sp+eps) -> S, chol(Sq+eps) -> Pr ----
#pragma unroll
  for (int q = 0; q < 8; ++q) {
    int e = e0 + q;
    float dg = ((e >> 4) == (e & 15)) ? EPS_KL : 0.f;
    S[e] = Sigma_prior[node * 256 + e] + dg;
  }
  wsync();
  chol16(S, lane);
#pragma unroll
  for (int q = 0; q < 8; ++q) {
    int e = e0 + q;
    float dg = ((e >> 4) == (e & 15)) ? EPS_KL : 0.f;
    Pr[e] = Sigma[node * 256 + e] + dg;
  }
  wsync();
  chol16(Pr, lane);

  // trace(Sp_inv * Sq): L=S, forward then in-place backward solve, lanes=cols.
  if (lane < 16) {
    int cidx = lane;
    for (int r = 0; r < 16; ++r) {
      float s = Sigma[node * 256 + r * 16 + cidx] + ((r == cidx) ? EPS_KL : 0.f);
      for (int p = 0; p < r; ++p) s -= S[r * 16 + p] * Mc[p * 16 + cidx];
      Mc[r * 16 + cidx] = s / S[r * 17];
    }
    for (int r = 15; r >= 0; --r) {
      float s = Mc[r * 16 + cidx];
      for (int p = r + 1; p < 16; ++p) s -= S[p * 16 + r] * Mc[p * 16 + cidx];
      Mc[r * 16 + cidx] = s / S[r * 17];
    }
  }
  wsync();
  float tz = (lane < 16) ? Mc[lane * 17] : 0.f;
  for (int m = 16; m >= 1; m >>= 1) tz += __shfl_xor(tz, m, 32);

  if (lane == 0) {
    float mah = 0.f;
    for (int r = 0; r < 16; ++r) {
      float s = mu_prior[node * 16 + r] - mu[node * 16 + r];
      for (int p = 0; p < r; ++p) s -= S[r * 16 + p] * scr[p];
      float vv = s / S[r * 17];
      scr[r] = vv;
      mah += vv * vv;
    }
    float ldp = 0.f, ldq = 0.f;
    for (int r = 0; r < 16; ++r) {
      ldp += logf(S[r * 17]);
      ldq += logf(Pr[r * 17]);
    }
    float kl = 0.5f * (tz + mah - 16.f + 2.f * (ldp - ldq));
    atomicAdd(&out[bb], kl);  // ALPHA = 1
  }
}

// =====================================================================
// Kernel 2: alignment pairs. 1 wave per (b,i,j), 4 waves/block,
// wave-private LDS (no block barriers).
// =====================================================================
__global__ void __launch_bounds__(128)
align_kernel(const float* __restrict__ mu, const float* __restrict__ Sigma,
             const float* __restrict__ beta, const float* __restrict__ wsEp,
             const float* __restrict__ wsEn, const float* __restrict__ wsLdq,
             float* __restrict__ out) {
  __shared__ float lds[4 * NODE_LDS];
  const int wid = threadIdx.x >> 5;
  const int lane = threadIdx.x & 31;
  float* Om = lds + wid * NODE_LDS;  // Omega
  float* Tm = Om + 256;              // Omega*Sigma_j
  float* St = Tm + 256;              // Sigma_t -> chol L
  float* Y = St + 256;               // solve buffer
  float* scr = Y + 256;              // 32-float scratch

  const int pair = blockIdx.x * 4 + wid;  // 0..73727
  const int b = pair / (Nc * Nc);
  const int rem = pair - b * Nc * Nc;
  const int i = rem / Nc, j = rem - (rem / Nc) * Nc;
  const int ni = b * Nc + i, nj = b * Nc + j;

  // Omega = Ep_i * En_j
  v8f c = mm16(wsEp + ni * 256, wsEn + nj * 256, lane);
  store_c(c, Om, lane);
  wsync();
  // tmp = Omega * Sigma_j
  c = mm16(Om, Sigma + nj * 256, lane);
  store_c(c, Tm, lane);
  wsync();
  // Sigma_t = tmp * Omega^T  (+ eps I)
  c = mm16_bt(Tm, Om, lane);
  {
    int h = (lane >> 4) & 1, n = lane & 15;
#pragma unroll
    for (int r = 0; r < 8; ++r) {
      int row = r + 8 * h;
      St[row * 16 + n] = c[r] + ((row == n) ? EPS_AL : 0.f);
    }
  }
  wsync();
  chol16(St, lane);  // St := L_p (lower)

  // tr = trace(L^-T L^-1 (Sigma_i + eps I)); lanes = columns.
  const float* Si = Sigma + ni * 256;
  if (lane < 16) {
    int cidx = lane;
    for (int r = 0; r < 16; ++r) {
      float s = Si[r * 16 + cidx] + ((r == cidx) ? EPS_AL : 0.f);
      for (int p = 0; p < r; ++p) s -= St[r * 16 + p] * Y[p * 16 + cidx];
      Y[r * 16 + cidx] = s / St[r * 17];
    }
    for (int r = 15; r >= 0; --r) {
      float s = Y[r * 16 + cidx];
      for (int p = r + 1; p < 16; ++p) s -= St[p * 16 + r] * Y[p * 16 + cidx];
      Y[r * 16 + cidx] = s / St[r * 17];
    }
  }
  wsync();
  float tz = (lane < 16) ? Y[lane * 17] : 0.f;
  for (int m = 16; m >= 1; m >>= 1) tz += __shfl_xor(tz, m, 32);

  // d2 = Omega*mu_j - mu_i
  if (lane < 16) {
    float s = 0.f;
    for (int l = 0; l < 16; ++l) s += Om[lane * 16 + l] * mu[nj * 16 + l];
    scr[lane] = s - mu[ni * 16 + lane];
  }
  wsync();

  if (lane == 0) {
    float mah = 0.f;
    for (int r = 0; r < 16; ++r) {
      float s = scr[r];
      for (int p = 0; p < r; ++p) s -= St[r * 16 + p] * scr[16 + p];
      float vv = s / St[r * 17];
      scr[16 + r] = vv;
      mah += vv * vv;
    }
    float ldp = 0.f;
    for (int r = 0; r < 16; ++r) ldp += logf(St[r * 17] + EPS_AL);
    ldp *= 2.f;
    float kl = 0.5f * (tz + mah - 16.f + ldp - wsLdq[ni]);
    kl = fmaxf(kl, 0.f);
    if (i != j)
      atomicAdd(&out[b], beta[b * Nc * Nc + i * Nc + j] * kl);  // LAMBDA = 1
  }
}

// =====================================================================
// Kernel 3: CE term. 8 waves per 16-row tile; each wave scans a V-slice
// with WMMA + online logsumexp, partials merged in LDS (cross-wave ->
// real __syncthreads, uniform trip counts).
// =====================================================================
#define CEW 8
__global__ void __launch_bounds__(32 * CEW)
ce_kernel(const float* __restrict__ mu, const float* __restrict__ W,
          const int* __restrict__ targets, float* __restrict__ out) {
  __shared__ float pm[CEW * 16], ps[CEW * 16], pt[CEW * 16];
  const int wid = threadIdx.x >> 5;
  const int lane = threadIdx.x & 31;
  const int row0 = blockIdx.x * 16;  // flattened b*N+n
  const int b = row0 / Nc;
  const int h = (lane >> 4) & 1, n = lane & 15;

  const float* Am = mu + row0 * 16;
  v2f a0 = frag_a_rm(Am, 0, lane), a1 = frag_a_rm(Am, 4, lane);
  v2f a2 = frag_a_rm(Am, 8, lane), a3 = frag_a_rm(Am, 12, lane);

  const int chunks_per_wave = (Vc / 16) / CEW;  // 250
  const int c0 = wid * chunks_per_wave, c1 = c0 + chunks_per_wave;

  float run_m[8], run_s[8], tlog[8];
  int tgt[8];
#pragma unroll
  for (int r = 0; r < 8; ++r) {
    run_m[r] = -3.0e38f;
    run_s[r] = 0.f;
    tlog[r] = 0.f;
    int t = targets[row0 + r + 8 * h];
    tgt[r] = (t < 0) ? 0 : ((t > Vc - 1) ? (Vc - 1) : t);
  }

  for (int ch = c0; ch < c1; ++ch) {
    int v0 = ch * 16;
    // B[k][col] = W[(v0+col)*16 + k]
    const float* Wr = W + (v0 + n) * 16;
    __builtin_prefetch((const void*)(Wr + 256), 0, 1);  // next chunk's row
    v2f b0, b1, b2, b3;
    b0.x = Wr[0 + 2 * h];  b0.y = Wr[1 + 2 * h];
    b1.x = Wr[4 + 2 * h];  b1.y = Wr[5 + 2 * h];
    b2.x = Wr[8 + 2 * h];  b2.y = Wr[9 + 2 * h];
    b3.x = Wr[12 + 2 * h]; b3.y = Wr[13 + 2 * h];
    v8f c = V8Z;
    c = wmma4(a0, b0, c);
    c = wmma4(a1, b1, c);
    c = wmma4(a2, b2, c);
    c = wmma4(a3, b3, c);
#pragma unroll
    for (int r = 0; r < 8; ++r) {
      float x = c[r];
      float cm = x;
      for (int m = 8; m >= 1; m >>= 1) cm = fmaxf(cm, __shfl_xor(cm, m, 32));
      float mn = fmaxf(run_m[r], cm);
      float p = expf(x - mn);
      float cs = p;
      for (int m = 8; m >= 1; m >>= 1) cs += __shfl_xor(cs, m, 32);
      run_s[r] = run_s[r] * expf(run_m[r] - mn) + cs;
      run_m[r] = mn;
      int t = tgt[r];
      float tl = ((t >= v0) && (t < v0 + 16) && (n == (t - v0))) ? x : 0.f;
      for (int m = 8; m >= 1; m >>= 1) tl += __shfl_xor(tl, m, 32);
      tlog[r] += tl;
    }
  }

  // publish per-wave partials (lanes 0 and 16 own 8 rows each)
  if (n == 0) {
#pragma unroll
    for (int r = 0; r < 8; ++r) {
      int row = r + 8 * h;
      pm[wid * 16 + row] = run_m[r];
      ps[wid * 16 + row] = run_s[r];
      pt[wid * 16 + row] = tlog[r];
    }
  }
  __syncthreads();

  // wave 0 merges the 8 partial logsumexps per row and commits.
  if (wid == 0 && lane < 16) {
    int row = lane;
    float M = -3.0e38f;
    for (int w = 0; w < CEW; ++w) M = fmaxf(M, pm[w * 16 + row]);
    float S = 0.f, T = 0.f;
    for (int w = 0; w < CEW; ++w) {
      S += ps[w * 16 + row] * expf(pm[w * 16 + row] - M);
      T += pt[w * 16 + row];
    }
    int t = targets[row0 + row];
    float nll = (t == -100) ? 0.f : (M + logf(S) - T);
    for (int m = 8; m >= 1; m >>= 1) nll += __shfl_xor(nll, m, 32);
    if (lane == 0) atomicAdd(&out[b], nll);
  }
}

__global__ void zero_out_kernel(float* out, int n) {
  int i = blockIdx.x * blockDim.x + threadIdx.x;
  if (i < n) out[i] = 0.f;
}

extern "C" void kernel_launch(void* const* d_in, const int* in_sizes, int n_in,
                              void* d_out, int out_size, void* d_ws,
                              size_t ws_size, hipStream_t stream) {
  (void)in_sizes; (void)n_in; (void)ws_size;
  const float* mu = (const float*)d_in[0];
  const float* Sigma = (const float*)d_in[1];
  const float* phi = (const float*)d_in[2];
  const float* mu_prior = (const float*)d_in[3];
  const float* Sigma_prior = (const float*)d_in[4];
  const float* beta = (const float*)d_in[5];
  const float* W_out = (const float*)d_in[6];
  const float* generators = (const float*)d_in[7];
  const int* targets = (const int*)d_in[8];
  float* out = (float*)d_out;

  float* ws = (float*)d_ws;
  float* wsEp = ws;                       // B*N*256
  float* wsEn = wsEp + Bc * Nc * 256;     // B*N*256
  float* wsLdq = wsEn + Bc * Nc * 256;    // B*N

  zero_out_kernel<<<1, 32, 0, stream>>>(out, out_size);
  node_kernel<<<(Bc * Nc) / 4, 128, 0, stream>>>(
      mu, Sigma, phi, mu_prior, Sigma_prior, generators, wsEp, wsEn, wsLdq, out);
  align_kernel<<<(Bc * Nc * Nc) / 4, 128, 0, stream>>>(
      mu, Sigma, beta, wsEp, wsEn, wsLdq, out);
  ce_kernel<<<(Bc * Nc) / 16, 32 * CEW, 0, stream>>>(mu, W_out, targets, out);
}